// CRF_1099511627978
// MI455X (gfx1250) — compile-verified
//
#include <hip/hip_runtime.h>

typedef float v2f __attribute__((ext_vector_type(2)));
typedef float v8f __attribute__((ext_vector_type(8)));

#define Bn 128
#define Ln 512
#define Hn 512
#define Tn 24
#define Mn (Bn * Ln)          // 65536 GEMM rows
#define START_TAG 22
#define STOP_TAG 23
#define NEGV (-10000.0f)

// ---------------------------------------------------------------------------
// Phase 1: emission GEMM  emit[m, t] = sum_h F[m,h] * W[t,h] + b[t]
// One wave computes a 16(M) x 32(N, padded from 24) tile via two f32 WMMAs
// per K-quad. FP32 WMMA: the GEMM is HBM-bound (12 FLOP/byte on 134 MB of
// f32 features), so full precision costs nothing.
// A 16x4 f32 layout: lane holds row m=lane%16, K = (lane<16?0:2)+{0,1}.
// B  4x16 f32 layout: lane holds col n=lane%16, same K split. Both are
// plain 8-byte contiguous loads (partner lanes share the same 16B line).
// ---------------------------------------------------------------------------
__global__ __launch_bounds__(256)
void crf_emit_gemm(const float* __restrict__ F,
                   const float* __restrict__ W,
                   const float* __restrict__ bias,
                   float* __restrict__ emit)
{
    const int lane = threadIdx.x & 31;
    const int wave = threadIdx.x >> 5;
    const int m0   = (blockIdx.x * 8 + wave) * 16;
    const int half = lane >> 4;        // K-half selector (0 -> K0,K1 ; 1 -> K2,K3)
    const int nloc = lane & 15;
    const int mrow = m0 + nloc;

    const int n1  = 16 + nloc;
    const int n1c = (n1 < Tn) ? n1 : (Tn - 1);   // clamp padded B rows

    const float* ap  = F + (size_t)mrow * Hn + half * 2;
    const float* bp0 = W + (size_t)nloc * Hn + half * 2;
    const float* bp1 = W + (size_t)n1c  * Hn + half * 2;

    v8f c0 = {};
    v8f c1 = {};
#pragma unroll 4
    for (int k = 0; k < Hn; k += 4) {
        const float2 av  = *(const float2*)(ap  + k);
        const float2 bv0 = *(const float2*)(bp0 + k);
        const float2 bv1 = *(const float2*)(bp1 + k);
        v2f a;  a[0]  = av.x;  a[1]  = av.y;
        v2f b0; b0[0] = bv0.x; b0[1] = bv0.y;
        v2f b1; b1[0] = bv1.x; b1[1] = bv1.y;
        c0 = __builtin_amdgcn_wmma_f32_16x16x4_f32(false, a, false, b0,
                                                   (short)0, c0, false, false);
        c1 = __builtin_amdgcn_wmma_f32_16x16x4_f32(false, a, false, b1,
                                                   (short)0, c1, false, false);
    }

    // C/D layout: VGPR r holds M = r (+8 for lanes 16-31), N = lane%16
    const float bias0 = bias[nloc];
    const float bias1 = bias[n1c];
    const int rbase = m0 + half * 8;

    float* o0 = emit + (size_t)rbase * Tn + nloc;
#pragma unroll
    for (int r = 0; r < 8; ++r)
        o0[(size_t)r * Tn] = c0[r] + bias0;

    if (n1 < Tn) {                     // single divergence region for tile 1
        float* o1 = emit + (size_t)rbase * Tn + n1;
#pragma unroll
        for (int r = 0; r < 8; ++r)
            o1[(size_t)r * Tn] = c1[r] + bias1;
    }
}

// ---------------------------------------------------------------------------
// Phase 2: CRF forward scan. One wave32 per batch element, lane j = tag j.
// fv broadcast through LDS (6 x ds_load_b128 / step). transitions row held
// in 24 VGPRs per lane. Early-exit at lengths[b] (fv frozen afterwards,
// matching the reference's where(t < len) semantics exactly).
// ---------------------------------------------------------------------------
__global__ __launch_bounds__(32)
void crf_scan(const float* __restrict__ emit,
              const float* __restrict__ trans,
              const int*   __restrict__ lengths,
              float*       __restrict__ out)
{
    __shared__ __align__(16) float fv[32];

    const int b = blockIdx.x;
    const int j = threadIdx.x;
    const bool active = (j < Tn);

    float trow[Tn];
    if (active) {
#pragma unroll
        for (int k = 0; k < Tn; ++k) trow[k] = trans[j * Tn + k];
    } else {
#pragma unroll
        for (int k = 0; k < Tn; ++k) trow[k] = NEGV;
    }

    fv[j] = (active && j == START_TAG) ? 0.0f : NEGV;

    const int len = lengths[b];
    const float* ep = emit + (size_t)b * Ln * Tn + j;

    for (int t = 0; t < len; ++t) {
        __syncthreads();                       // single-wave WG: S_NOP-class
        float f[Tn];
#pragma unroll
        for (int q = 0; q < Tn / 4; ++q) {     // 6 x ds_load_b128
            const float4 v = *(const float4*)&fv[q * 4];
            f[q * 4 + 0] = v.x; f[q * 4 + 1] = v.y;
            f[q * 4 + 2] = v.z; f[q * 4 + 3] = v.w;
        }

        float s[Tn];
#pragma unroll
        for (int k = 0; k < Tn; ++k) s[k] = trow[k] + f[k];

        float m01 = s[0];
#pragma unroll
        for (int k = 1; k < Tn; ++k) m01 = fmaxf(m01, s[k]);

        float sum = 0.0f;
#pragma unroll
        for (int k = 0; k < Tn; ++k) sum += __expf(s[k] - m01);

        const float e = active ? ep[(size_t)t * Tn] : NEGV;
        if ((t + 1) < len)                     // wave-uniform -> scalar branch
            __builtin_prefetch(ep + (size_t)(t + 1) * Tn, 0, 1); // global_prefetch_b8

        const float nv = e + m01 + __logf(sum);

        __syncthreads();
        fv[j] = active ? nv : NEGV;
    }
    __syncthreads();

    if (j == 0) {
        float s[Tn];
#pragma unroll
        for (int k = 0; k < Tn; ++k) s[k] = fv[k] + trans[STOP_TAG * Tn + k];
        float m = s[0];
#pragma unroll
        for (int k = 1; k < Tn; ++k) m = fmaxf(m, s[k]);
        float sum = 0.0f;
#pragma unroll
        for (int k = 0; k < Tn; ++k) sum += __expf(s[k] - m);
        out[b] = m + __logf(sum);
    }
}

// ---------------------------------------------------------------------------
extern "C" void kernel_launch(void* const* d_in, const int* in_sizes, int n_in,
                              void* d_out, int out_size, void* d_ws, size_t ws_size,
                              hipStream_t stream)
{
    const float* features    = (const float*)d_in[0];   // [B, L, H]
    const float* emission_w  = (const float*)d_in[1];   // [T, H]
    const float* emission_b  = (const float*)d_in[2];   // [T]
    const float* transitions = (const float*)d_in[3];   // [T, T]
    const int*   lengths     = (const int*)d_in[4];     // [B]
    float*       out         = (float*)d_out;           // [B]

    float* emit = (float*)d_ws;                         // [B*L, T] = 6.3 MB

    // 65536/16 M-tiles / 8 waves per block = 512 blocks
    crf_emit_gemm<<<Mn / 16 / 8, 256, 0, stream>>>(features, emission_w,
                                                   emission_b, emit);
    crf_scan<<<Bn, 32, 0, stream>>>(emit, transitions, lengths, out);
}